// SimpleRNN_28217935135279
// MI455X (gfx1250) — compile-verified
//
#include <hip/hip_runtime.h>
#include <stdint.h>

// ---------------- feature probes (device pass only) ----------------
#if defined(__AMDGCN__)
#  if __has_builtin(__builtin_amdgcn_global_load_async_to_lds_b128)
#    define HAVE_ASYNC_LDS 1
#  else
#    warning "CDNA5 async-to-LDS builtin (__builtin_amdgcn_global_load_async_to_lds_b128) NOT found - using fallback loads"
#    define HAVE_ASYNC_LDS 0
#  endif
#  if __has_builtin(__builtin_amdgcn_tanhf)
#    define TANHF(v) __builtin_amdgcn_tanhf(v)
#  elif __has_builtin(__builtin_amdgcn_tanh_f32)
#    define TANHF(v) __builtin_amdgcn_tanh_f32(v)
#  else
#    warning "no __builtin_amdgcn_tanhf / __builtin_amdgcn_tanh_f32 - falling back to tanhf()"
#    define TANHF(v) tanhf(v)
#  endif
#else
#  define HAVE_ASYNC_LDS 0
#  define TANHF(v) tanhf(v)
#endif

#if HAVE_ASYNC_LDS
#  if __has_builtin(__builtin_amdgcn_s_wait_asynccnt)
#    define WAIT_ASYNC(n) __builtin_amdgcn_s_wait_asynccnt(n)
#  else
#    define WAIT_ASYNC(n) asm volatile("s_wait_asynccnt " #n ::: "memory")
#  endif
#endif

// ---------------- tuning constants ----------------
#define ROWS_PER_BLOCK 32      // one wave32, one row per lane
#define CHUNK          128     // floats of each row staged per chunk (512 B/row)
#define LDS_STRIDE     132     // padded row stride in floats (16B aligned, conflict-light)

#if HAVE_ASYNC_LDS
// b128 async copy moves one int4 per lane; builtin params are v4i pointers.
typedef int v4i __attribute__((ext_vector_type(4)));
typedef __attribute__((address_space(1))) v4i gv4i;   // global src
typedef __attribute__((address_space(3))) v4i lv4i;   // LDS dst

__device__ __forceinline__ void async_copy16(const float* g, float* l) {
    // global_load_async_to_lds_b128: per-lane 16B global -> LDS, tracked by ASYNCcnt
    __builtin_amdgcn_global_load_async_to_lds_b128(
        (gv4i*)(uintptr_t)g,
        (lv4i*)(unsigned)(uintptr_t)l,
        0 /*imm offset*/, 0 /*cpol*/);
}

// Stage a [32 rows x CHUNK cols] tile. 32 lanes x 16B cover one 128-float row
// per instruction (fully coalesced 512B transactions); loop over the 32 rows.
__device__ __forceinline__ void issue_tile(const float* gbase_lane, float* lds_lane, int T) {
#pragma unroll
    for (int r = 0; r < ROWS_PER_BLOCK; ++r) {
        async_copy16(gbase_lane + (size_t)r * T, lds_lane + r * LDS_STRIDE);
    }
}
#endif

__global__ __launch_bounds__(ROWS_PER_BLOCK)
void SimpleRNN_scan_kernel(const float* __restrict__ x,
                           const float* __restrict__ w_ih,
                           const float* __restrict__ w_hh,
                           const float* __restrict__ b_ih,
                           const float* __restrict__ b_hh,
                           float* __restrict__ out,
                           int T) {
    const int lane = threadIdx.x;                       // 0..31
    const int row0 = blockIdx.x * ROWS_PER_BLOCK;
    const int row  = row0 + lane;

    const float a  = w_ih[0];
    const float bb = w_hh[0];
    const float c  = b_ih[0] + b_hh[0];

    float h = 0.0f;

#if HAVE_ASYNC_LDS
    __shared__ float buf[2][ROWS_PER_BLOCK * LDS_STRIDE];

    const int nchunk = T / CHUNK;
    // lane-resolved base pointers: lane covers cols [lane*4, lane*4+4) of every row
    const float* gbase = x + (size_t)row0 * T + lane * 4;
    float* l0 = &buf[0][lane * 4];
    float* l1 = &buf[1][lane * 4];

    issue_tile(gbase, l0, T);                           // prologue: chunk 0 in flight

    for (int k = 0; k < nchunk; ++k) {
        if (k + 1 < nchunk) {
            // double-buffer: prefetch chunk k+1 while we consume chunk k
            issue_tile(gbase + (size_t)(k + 1) * CHUNK, ((k + 1) & 1) ? l1 : l0, T);
            WAIT_ASYNC(32);                             // chunk k's 32 ops retired (in-order)
        } else {
            WAIT_ASYNC(0);
        }
        asm volatile("" ::: "memory");                  // order ds reads after the wait

        const float* r = &buf[k & 1][lane * LDS_STRIDE];
        // Recurrence: u = a*x + c is independent (scheduled ahead);
        // dependent chain per step is exactly fma + v_tanh_f32.
#pragma unroll
        for (int t = 0; t < CHUNK; t += 4) {
            float4 v = *(const float4*)(r + t);         // ds_load_b128
            h = TANHF(fmaf(h, bb, fmaf(v.x, a, c)));
            h = TANHF(fmaf(h, bb, fmaf(v.y, a, c)));
            h = TANHF(fmaf(h, bb, fmaf(v.z, a, c)));
            h = TANHF(fmaf(h, bb, fmaf(v.w, a, c)));
        }
    }
#else
    // Fallback: direct (uncoalesced) per-row streaming, still correct.
    const float* xr = x + (size_t)row * T;
    for (int t = 0; t < T; t += 4) {
        float4 v = *(const float4*)(xr + t);
        h = TANHF(fmaf(h, bb, fmaf(v.x, a, c)));
        h = TANHF(fmaf(h, bb, fmaf(v.y, a, c)));
        h = TANHF(fmaf(h, bb, fmaf(v.z, a, c)));
        h = TANHF(fmaf(h, bb, fmaf(v.w, a, c)));
    }
#endif

    out[row] = h;
}

extern "C" void kernel_launch(void* const* d_in, const int* in_sizes, int n_in,
                              void* d_out, int out_size, void* d_ws, size_t ws_size,
                              hipStream_t stream) {
    (void)n_in; (void)d_ws; (void)ws_size;
    const float* x    = (const float*)d_in[0];
    const float* w_ih = (const float*)d_in[1];
    const float* w_hh = (const float*)d_in[2];
    const float* b_ih = (const float*)d_in[3];
    const float* b_hh = (const float*)d_in[4];
    float* out = (float*)d_out;

    const int B = out_size;                 // 8192
    const int T = in_sizes[0] / B;          // 4096
    const int nblocks = B / ROWS_PER_BLOCK; // 256 single-wave workgroups -> spread over WGPs

    SimpleRNN_scan_kernel<<<dim3(nblocks), dim3(ROWS_PER_BLOCK), 0, stream>>>(
        x, w_ih, w_hh, b_ih, b_hh, out, T);
}